// NLL_loss_v1_84430467105163
// MI455X (gfx1250) — compile-verified
//
#include <hip/hip_runtime.h>

#define TPB   256
#define EPT   16
#define CHUNK (TPB * EPT)   // 4096 elements per block
#define TPAD  0x7fffffff

typedef float v2f __attribute__((ext_vector_type(2)));
typedef float v8f __attribute__((ext_vector_type(8)));

// ---- CDNA5 async global->LDS copies (ASYNCcnt-tracked), per CDNA5 ISA ----
__device__ __forceinline__ void async_ld_b32(void* lds_ptr, const void* gptr) {
  unsigned lds_off = (unsigned)(unsigned long long)lds_ptr;  // low 32 bits = LDS byte offset
  asm volatile("global_load_async_to_lds_b32 %0, %1, off"
               :: "v"(lds_off), "v"(gptr) : "memory");
}
__device__ __forceinline__ void async_ld_b128(void* lds_ptr, const void* gptr) {
  unsigned lds_off = (unsigned)(unsigned long long)lds_ptr;
  asm volatile("global_load_async_to_lds_b128 %0, %1, off"
               :: "v"(lds_off), "v"(gptr) : "memory");
}
__device__ __forceinline__ void wait_async0() {
  asm volatile("s_wait_asynccnt 0" ::: "memory");
}

// K1: per-block partial sums: P[b] = sum exp(risk), Q[b] = sum event*risk
__global__ void k1_partials(const float* __restrict__ risk, const float* __restrict__ ev,
                            float* __restrict__ P, float* __restrict__ Q, long n) {
  __shared__ float r1[TPB], r2[TPB];
  const int t = threadIdx.x;
  const long base = (long)blockIdx.x * CHUNK;
  float s = 0.f, q = 0.f;
  if (base + CHUNK <= n) {                      // fast path: b128 loads
    const float4* r4 = (const float4*)(risk + base);
    const float4* e4 = (const float4*)(ev + base);
#pragma unroll
    for (int k = 0; k < EPT / 4; ++k) {
      float4 r = r4[k * TPB + t];
      float4 e = e4[k * TPB + t];
      s += expf(r.x) + expf(r.y) + expf(r.z) + expf(r.w);
      q += e.x * r.x + e.y * r.y + e.z * r.z + e.w * r.w;
    }
  } else {                                      // tail block: guarded scalar
    for (int k = 0; k < EPT; ++k) {
      long i = base + (long)k * TPB + t;
      if (i < n) { float r = risk[i]; s += expf(r); q += ev[i] * r; }
    }
  }
  r1[t] = s; r2[t] = q; __syncthreads();
  for (int off = TPB / 2; off > 0; off >>= 1) {
    if (t < off) { r1[t] += r1[t + off]; r2[t] += r2[t + off]; }
    __syncthreads();
  }
  if (t == 0) { P[blockIdx.x] = r1[0]; Q[blockIdx.x] = r2[0]; }
}

// K2 (1 block): exclusive scan of P -> O ; reduce Q -> *SQ. Deterministic.
__global__ void k2_scan(const float* __restrict__ P, const float* __restrict__ Q,
                        float* __restrict__ O, float* __restrict__ SQ, int nb) {
  __shared__ float s[TPB];
  __shared__ float carry;
  const int t = threadIdx.x;
  if (t == 0) carry = 0.f;
  __syncthreads();
  const int nch = (nb + TPB - 1) / TPB;
  for (int c = 0; c < nch; ++c) {
    int idx = c * TPB + t;
    float v = (idx < nb) ? P[idx] : 0.f;
    s[t] = v; __syncthreads();
    for (int off = 1; off < TPB; off <<= 1) {        // Hillis-Steele inclusive scan
      float x = s[t]; float a = (t >= off) ? s[t - off] : 0.f;
      __syncthreads(); s[t] = x + a; __syncthreads();
    }
    float incl = s[t];
    float b0 = carry;
    if (idx < nb) O[idx] = b0 + incl - v;            // exclusive offset
    __syncthreads();
    if (t == TPB - 1) carry = b0 + incl;
    __syncthreads();
  }
  float q = 0.f;
  for (int idx = t; idx < nb; idx += TPB) q += Q[idx];
  s[t] = q; __syncthreads();
  for (int off = TPB / 2; off > 0; off >>= 1) { if (t < off) s[t] += s[t + off]; __syncthreads(); }
  if (t == 0) *SQ = s[0];
}

// K3: block-local inclusive scan of exp(risk) + global offset; write cumsum at
// tie-group ends into Ct[time] (single writer per distinct time -> no atomics).
// Stages risk/time through LDS with CDNA5 async b128 global->LDS loads.
__global__ void k3_groupends(const float* __restrict__ risk, const int* __restrict__ tim,
                             const float* __restrict__ O, float* __restrict__ Ct, long n) {
  __shared__ __align__(16) float vals[CHUNK];
  __shared__ __align__(16) int   tms[CHUNK + 4];
  __shared__ float sscan[TPB];
  const int t = threadIdx.x;
  const long base = (long)blockIdx.x * CHUNK;
  if (base + CHUNK <= n) {                      // fast path: async b128 (16B/lane)
#pragma unroll
    for (int k = 0; k < EPT / 4; ++k) {
      int e0 = (k * TPB + t) * 4;
      async_ld_b128(&vals[e0], risk + base + e0);
      async_ld_b128(&tms[e0],  tim  + base + e0);
    }
  } else {                                      // tail block: guarded scalar fill
    for (int k = 0; k < EPT; ++k) {
      int idx = k * TPB + t;
      long i = base + idx;
      if (i < n) { async_ld_b32(&vals[idx], risk + i); async_ld_b32(&tms[idx], tim + i); }
      else       { vals[idx] = -__builtin_inff(); tms[idx] = TPAD; }  // expf(-inf)==0
    }
  }
  if (t == 0) tms[CHUNK] = (base + CHUNK < n) ? tim[base + CHUNK] : TPAD;
  wait_async0();
  __syncthreads();

  float pref[EPT];
  float run = 0.f;
  const int s0 = t * EPT;
#pragma unroll
  for (int j = 0; j < EPT; ++j) { run += expf(vals[s0 + j]); pref[j] = run; }

  sscan[t] = run; __syncthreads();
  for (int off = 1; off < TPB; off <<= 1) {
    float x = sscan[t]; float a = (t >= off) ? sscan[t - off] : 0.f;
    __syncthreads(); sscan[t] = x + a; __syncthreads();
  }
  const float cbase = O[blockIdx.x] + (sscan[t] - run);  // global exclusive prefix for this thread
#pragma unroll
  for (int j = 0; j < EPT; ++j) {
    int tc = tms[s0 + j], tn = tms[s0 + j + 1];
    if (tc != tn && tc != TPAD) Ct[tc] = cbase + pref[j];
  }
}

// K4: partial sums of event * log(Ct[time]) -> R[b]. Ct is L2-resident (400KB);
// time sorted => consecutive lanes hit same Ct entry (broadcast).
__global__ void k4_logsum(const float* __restrict__ ev, const int* __restrict__ tim,
                          const float* __restrict__ Ct, float* __restrict__ R, long n) {
  __shared__ float r1[TPB];
  const int t = threadIdx.x;
  const long base = (long)blockIdx.x * CHUNK;
  float s = 0.f;
  if (base + CHUNK <= n) {                      // fast path: b128 loads
    const float4* e4 = (const float4*)(ev + base);
    const int4*   t4 = (const int4*)(tim + base);
#pragma unroll
    for (int k = 0; k < EPT / 4; ++k) {
      float4 e = e4[k * TPB + t];
      int4   x = t4[k * TPB + t];
      s += e.x * logf(Ct[x.x]) + e.y * logf(Ct[x.y])
         + e.z * logf(Ct[x.z]) + e.w * logf(Ct[x.w]);
    }
  } else {
    for (int k = 0; k < EPT; ++k) {
      long i = base + (long)k * TPB + t;
      if (i < n) s += ev[i] * logf(Ct[tim[i]]);
    }
  }
  r1[t] = s; __syncthreads();
  for (int off = TPB / 2; off > 0; off >>= 1) { if (t < off) r1[t] += r1[t + off]; __syncthreads(); }
  if (t == 0) R[blockIdx.x] = r1[0];
}

// K5 (1 wave): WMMA-based reduction of R. A = all-ones 16x4 =>
// grand(D) = 16*grand(B) + grand(C); layout-independent, exact f32.
__global__ void k5_final(const float* __restrict__ R, const float* __restrict__ SQ,
                         float* __restrict__ out, int nb, float fn) {
  const int t = threadIdx.x;           // launched with exactly 32 threads (EXEC all-1s)
  v2f a; a.x = 1.f; a.y = 1.f;
  v8f c = {0.f, 0.f, 0.f, 0.f, 0.f, 0.f, 0.f, 0.f};
  const v2f* R2 = (const v2f*)R;       // R is 16B-aligned in ws
  const int full = nb >> 6;            // iterations with all 32 lanes' values valid
  for (int it = 0; it < full; ++it) {  // unguarded b64 loads, clean WMMA loop
    v2f b = R2[it * 32 + t];
    c = __builtin_amdgcn_wmma_f32_16x16x4_f32(false, a, false, b, (short)0, c, false, false);
  }
  if (nb & 63) {                       // uniform branch: EXEC stays full at the WMMA
    int i0 = full * 64 + t * 2;
    v2f b;
    b.x = (i0     < nb) ? R[i0]     : 0.f;
    b.y = (i0 + 1 < nb) ? R[i0 + 1] : 0.f;
    c = __builtin_amdgcn_wmma_f32_16x16x4_f32(false, a, false, b, (short)0, c, false, false);
  }
  float s = c[0] + c[1] + c[2] + c[3] + c[4] + c[5] + c[6] + c[7];
  for (int off = 16; off > 0; off >>= 1) s += __shfl_down(s, off, 32);
  if (t == 0) {
    float SR = s * 0.0625f;            // grand(C)/16 == sum(R), exact power-of-2 scale
    out[0] = -(SQ[0] - SR) / fn;
  }
}

extern "C" void kernel_launch(void* const* d_in, const int* in_sizes, int n_in,
                              void* d_out, int out_size, void* d_ws, size_t ws_size,
                              hipStream_t stream) {
  (void)n_in; (void)out_size; (void)ws_size;
  const float* risk = (const float*)d_in[0];
  const float* ev   = (const float*)d_in[1];
  const int*   tim  = (const int*)d_in[2];
  float* out = (float*)d_out;
  const long n = (long)in_sizes[0];
  const int nb = (int)((n + CHUNK - 1) / CHUNK);   // 4096 for N=16.7M

  float* ws = (float*)d_ws;
  float* P  = ws;            // cap 8192
  float* O  = ws + 8192;     // cap 8192
  float* Q  = ws + 16384;    // cap 8192
  float* R  = ws + 24576;    // cap 8192 (16B-aligned)
  float* SQ = ws + 32768;    // 1
  float* Ct = ws + 32832;    // 100000 entries (T_MAX), ~400KB, L2-resident

  k1_partials <<<nb, TPB, 0, stream>>>(risk, ev, P, Q, n);
  k2_scan     <<<1,  TPB, 0, stream>>>(P, Q, O, SQ, nb);
  k3_groupends<<<nb, TPB, 0, stream>>>(risk, tim, O, Ct, n);
  k4_logsum   <<<nb, TPB, 0, stream>>>(ev, tim, Ct, R, n);
  k5_final    <<<1,  32,  0, stream>>>(R, SQ, out, nb, (float)n);
}